// BaseBLModel_13735305412889
// MI455X (gfx1250) — compile-verified
//
#include <hip/hip_runtime.h>
#include <hip/hip_bf16.h>
#include <math.h>

// ---------------------------------------------------------------------------
// Black-Litterman posterior, rewritten to avoid both explicit 128x128
// inversions:
//   A1 = tau*Sigma + 1e-6 I           (SPD)
//   D  = diag(p^2/omega) + 1e-6 I     (diagonal, positive)
//   mu = (inv(A1)+D)^-1 (inv(A1) pi + d)        [reference]
//      = (I + A1 D)^-1 (pi + A1 d)              [push A1 through]
//      = D^-1/2 (I + D^1/2 A1 D^1/2)^-1 D^1/2 (pi + A1 d)   [symmetrize, SPD]
// => one Cholesky solve per batch element instead of two explicit inverses
//    (~12x fewer FLOPs); kernel is HBM-bound on reading Sigma (134 MB ~ 6us).
// Sigma staging uses the CDNA5 Tensor Data Mover (tensor_load_to_lds) with
// hardware row padding (128 DWORD interval + 1 DWORD pad -> LDS stride 129,
// bank-conflict-free column access), tracked with TENSORcnt.
// ---------------------------------------------------------------------------

typedef __attribute__((ext_vector_type(2))) float v2f;
typedef __attribute__((ext_vector_type(8))) float v8f;
typedef __attribute__((ext_vector_type(4))) unsigned int u32x4;
typedef __attribute__((ext_vector_type(8))) int i32x8;
typedef __attribute__((ext_vector_type(4))) int i32x4;

#define K_TAU 0.05f
#define NN 128
#define HH 512
#define LDA 129   // padded LDS stride: 129 % 64 == 1 -> conflict-free columns

// ---------------------------------------------------------------------------
// Kernel 1: fused projections.  One wave32 per 16(b) x 16(n) tile.
// Z{q,p,o} = hidden @ W{q,p,o}^T + b via V_WMMA_F32_16X16X4_F32 (A shared by
// the three accumulator chains).  Activations fused:
//   q = tanh(zq), p = sigmoid(zp), w = softplus(zo)+1e-6
//   td = (p/w)*q          (view term d = P^T Omega^-1 Q)
//   Dd = p*p/w + 1e-6     (diagonal D)
// Weights (768 KB) + hidden (4 MB) live in the 192 MB L2, so re-reads are
// cheap; no LDS staging needed here.
// ---------------------------------------------------------------------------
__global__ __launch_bounds__(32) void bl_proj_kernel(
    const float* __restrict__ hidden,
    const float* __restrict__ Wq, const float* __restrict__ bq,
    const float* __restrict__ Wp, const float* __restrict__ bp,
    const float* __restrict__ Wo, const float* __restrict__ bo,
    float* __restrict__ td, float* __restrict__ Dd)
{
    const int lane = threadIdx.x;          // 0..31
    const int bt   = blockIdx.x;           // batch tile (16 rows)
    const int nt   = blockIdx.y;           // output tile (16 cols)
    const int mm   = lane & 15;
    const int hh   = lane >> 4;

    const float* arow = hidden + (size_t)(bt * 16 + mm) * HH;  // A: M = lane&15
    const float* qrow = Wq + (size_t)(nt * 16 + mm) * HH;      // B^T row: N = lane&15
    const float* prow = Wp + (size_t)(nt * 16 + mm) * HH;
    const float* orow = Wo + (size_t)(nt * 16 + mm) * HH;

    v8f cq = {}; v8f cp = {}; v8f co = {};
    for (int k = 0; k < HH; k += 4) {      // WMMA K=4; per-lane K slots 2*hh+{0,1}
        const int ka = k + 2 * hh;
        const v2f a   = *(const v2f*)(arow + ka);
        const v2f bqv = *(const v2f*)(qrow + ka);
        const v2f bpv = *(const v2f*)(prow + ka);
        const v2f bov = *(const v2f*)(orow + ka);
        cq = __builtin_amdgcn_wmma_f32_16x16x4_f32(false, a, false, bqv, (short)0, cq, false, false);
        cp = __builtin_amdgcn_wmma_f32_16x16x4_f32(false, a, false, bpv, (short)0, cp, false, false);
        co = __builtin_amdgcn_wmma_f32_16x16x4_f32(false, a, false, bov, (short)0, co, false, false);
    }

    const int ncol = nt * 16 + mm;
    const float bqs = bq[ncol], bps = bp[ncol], bos = bo[ncol];

    #pragma unroll
    for (int v = 0; v < 8; ++v) {
        const int brow = bt * 16 + v + 8 * hh;       // C layout: M = v + 8*(lane>>4)
        const float zq = cq[v] + bqs;
        const float zp = cp[v] + bps;
        const float zo = co[v] + bos;
        const float q  = tanhf(zq);
        const float p  = 1.0f / (1.0f + expf(-zp));
        const float sp = fmaxf(zo, 0.0f) + log1pf(expf(-fabsf(zo)));   // softplus
        const float w  = sp + 1e-6f;
        const float ratio = p / w;
        td[(size_t)brow * NN + ncol] = ratio * q;
        Dd[(size_t)brow * NN + ncol] = p * ratio + 1e-6f;
    }
}

// ---------------------------------------------------------------------------
// Kernel 2: one workgroup (256 threads = 8 wave32) per batch element.
//  1. TDM: tensor_load_to_lds stages raw Sigma (128x128 f32) into LDS with
//     hardware padding -> stride 129; wave0 waits on TENSORcnt.
//  2. One fused LDS sweep: a1 = tau*s + eps*delta; accumulate A1*d matvec
//     partials; write back M = I + sqrt(D) A1 sqrt(D)   (SPD).
//  3. Blocked Cholesky, NB=16: panel factor in wave0 (wave32 lockstep,
//     barrier-free), rank-16 trailing updates as WMMA f32 16x16x4 tiles
//     distributed over all 8 waves (uniform control flow -> EXEC all-1s).
//  4. Triangular solves L z = rhs, L^T y = z in wave0 (barrier-free).
//  5. mu = y ./ sqrt(D)
// ---------------------------------------------------------------------------
__global__ __launch_bounds__(256) void bl_solve_kernel(
    const float* __restrict__ pi, const float* __restrict__ sigma,
    const float* __restrict__ td, const float* __restrict__ Dd,
    float* __restrict__ mu)
{
    extern __shared__ float M[];                     // 128 x 129 f32 (raw, then M)
    __shared__ float dvec[NN], sD[NN], bvec[NN], yvec[NN], red[256];

    const int tid  = threadIdx.x;
    const int wave = tid >> 5;
    const int lane = tid & 31;
    const int b    = blockIdx.x;
    const float* sig = sigma + (size_t)b * NN * NN;

    if (tid < NN) {
        dvec[tid] = td[(size_t)b * NN + tid];
        sD[tid]   = sqrtf(Dd[(size_t)b * NN + tid]);
    }

#if __has_builtin(__builtin_amdgcn_tensor_load_to_lds) && __has_builtin(__builtin_amdgcn_s_wait_tensorcnt)
    // ---- TDM staging: one descriptor, issued by wave0 only ----------------
    if (tid < 32) {
        const unsigned long long ga = (unsigned long long)(uintptr_t)sig;
        const unsigned int lds_base = (unsigned int)(uintptr_t)(void*)&M[0];

        u32x4 g0;
        g0[0] = 1u;                                    // count=1, user mode
        g0[1] = lds_base;                              // lds_addr (bytes)
        g0[2] = (unsigned int)ga;                      // global_addr[31:0]
        g0[3] = (unsigned int)((ga >> 32) & 0x1FFFFFFu) | (2u << 30);  // addr[56:32] | type=2

        i32x8 g1;
        g1[0] = (2 << 16)      // data_size = 4 bytes
              | (1 << 20)      // pad_enable (load only)
              | (6 << 22);     // pad_interval: 128 DWORDs
                               // pad_amount [31:25] = 0 -> 1 DWORD  => stride 129
        g1[1] = (int)((NN & 0xffff) << 16);                    // tensor_dim0 lo16
        g1[2] = (int)(((NN >> 16) & 0xffff) | ((NN & 0xffff) << 16)); // dim0 hi | dim1 lo
        g1[3] = (int)(((NN >> 16) & 0xffff) | (NN << 16));     // dim1 hi | tile_dim0
        g1[4] = NN;                                            // tile_dim1 | tile_dim2=0
        g1[5] = NN;                                            // tensor_dim0_stride lo32
        g1[6] = 0;
        g1[7] = 0;

        const i32x4 z4 = {0, 0, 0, 0};
#if __clang_major__ >= 23
        const i32x8 z8 = {0, 0, 0, 0, 0, 0, 0, 0};
        __builtin_amdgcn_tensor_load_to_lds(g0, g1, z4, z4, z8, 0);
#else
        __builtin_amdgcn_tensor_load_to_lds(g0, g1, z4, z4, 0);
#endif
        __builtin_amdgcn_s_wait_tensorcnt(0);
    }
#else
    // ---- fallback: stage raw Sigma via 16B vector loads -------------------
    for (int idx = tid * 4; idx < NN * NN; idx += 256 * 4) {
        const float4 v = *reinterpret_cast<const float4*>(sig + idx);
        const int r = idx >> 7, c = idx & 127;
        float* dst = &M[r * LDA + c];
        dst[0] = v.x; dst[1] = v.y; dst[2] = v.z; dst[3] = v.w;
    }
#endif
    __syncthreads();

    // ---- fused sweep: A1 matvec partials + M = I + sD A1 sD ---------------
    // Each element (r,c) is touched by exactly one thread -> RMW is race-free.
    {
        const int r = tid >> 1, h = tid & 1;
        const float sdr = sD[r];
        float s = 0.0f;
        #pragma unroll 8
        for (int c0 = 0; c0 < 64; ++c0) {
            const int c = h * 64 + c0;
            const float raw = M[r * LDA + c];
            const float a1  = K_TAU * raw + ((r == c) ? 1e-6f : 0.0f);
            s += a1 * dvec[c];
            float m = a1 * sdr * sD[c];
            if (r == c) m += 1.0f;
            M[r * LDA + c] = m;
        }
        red[tid] = s;
    }
    __syncthreads();
    if (tid < NN)
        bvec[tid] = (pi[(size_t)b * NN + tid] + red[2 * tid] + red[2 * tid + 1]) * sD[tid];
    __syncthreads();

    // ---- blocked Cholesky (lower), NB = 16 --------------------------------
    for (int kb = 0; kb < NN / 16; ++kb) {
        const int j0 = kb * 16;

        // Panel factorization: single wave, lockstep, no barriers needed.
        if (wave == 0) {
            for (int j = 0; j < 16; ++j) {
                const int jj = j0 + j;
                const float diag = M[jj * LDA + jj];
                const float linv = __frsqrt_rn(diag);
                if (lane == 0) M[jj * LDA + jj] = diag * linv;       // sqrt(diag)
                for (int i = jj + 1 + lane; i < NN; i += 32)
                    M[i * LDA + jj] *= linv;
                for (int t = jj + 1; t < j0 + 16; ++t) {
                    const float ltj = M[t * LDA + jj];
                    for (int i = t + lane; i < NN; i += 32)
                        M[i * LDA + t] -= M[i * LDA + jj] * ltj;
                }
            }
        }
        __syncthreads();

        // Trailing update: C(bi,bj) -= Lp(bi) * Lp(bj)^T, K=16 -> 4 wmma.
        const int nb_rem = (NN / 16) - 1 - kb;
        if (nb_rem > 0) {
            const int ntiles = nb_rem * (nb_rem + 1) / 2;
            const int mm = lane & 15, hh = lane >> 4;
            for (int t = wave; t < ntiles; t += 8) {     // uniform per wave -> EXEC all-1
                int bi = 0, tt = t;
                while (tt >= bi + 1) { tt -= bi + 1; ++bi; }
                const int bj = tt;
                const int r0 = (kb + 1 + bi) * 16;
                const int c0 = (kb + 1 + bj) * 16;
                v8f acc = {};
                #pragma unroll
                for (int kk = 0; kk < 16; kk += 4) {
                    v2f af, bf;
                    const int ks = j0 + kk + 2 * hh;
                    af.x = M[(r0 + mm) * LDA + ks];
                    af.y = M[(r0 + mm) * LDA + ks + 1];
                    bf.x = M[(c0 + mm) * LDA + ks];      // B[k][n] = L[c0+n][j0+k]
                    bf.y = M[(c0 + mm) * LDA + ks + 1];
                    acc = __builtin_amdgcn_wmma_f32_16x16x4_f32(
                        false, af, false, bf, (short)0, acc, false, false);
                }
                #pragma unroll
                for (int v = 0; v < 8; ++v)
                    M[(r0 + v + 8 * hh) * LDA + c0 + mm] -= acc[v];
            }
        }
        __syncthreads();
    }

    // ---- triangular solves in wave0 (lockstep, no barriers) ---------------
    if (wave == 0) {
        // forward: L z = rhs (in place in bvec)
        for (int j = 0; j < NN; ++j) {
            const float zj = bvec[j] / M[j * LDA + j];
            if (lane == 0) bvec[j] = zj;
            for (int i = j + 1 + lane; i < NN; i += 32)
                bvec[i] -= M[i * LDA + j] * zj;
        }
        // backward: L^T y = z
        for (int j = NN - 1; j >= 0; --j) {
            const float yj = bvec[j] / M[j * LDA + j];
            if (lane == 0) yvec[j] = yj;
            for (int i = lane; i < j; i += 32)
                bvec[i] -= M[j * LDA + i] * yj;          // row-major read: no conflicts
        }
    }
    __syncthreads();
    if (tid < NN) mu[(size_t)b * NN + tid] = yvec[tid] / sD[tid];
}

// ---------------------------------------------------------------------------
extern "C" void kernel_launch(void* const* d_in, const int* in_sizes, int n_in,
                              void* d_out, int out_size, void* d_ws, size_t ws_size,
                              hipStream_t stream) {
    (void)n_in; (void)out_size; (void)ws_size;
    const float* hidden = (const float*)d_in[0];
    const float* pi     = (const float*)d_in[1];
    const float* sigma  = (const float*)d_in[2];
    const float* Wq     = (const float*)d_in[3];
    const float* bq     = (const float*)d_in[4];
    const float* Wp     = (const float*)d_in[5];
    const float* bp     = (const float*)d_in[6];
    const float* Wo     = (const float*)d_in[7];
    const float* bo     = (const float*)d_in[8];
    float* mu = (float*)d_out;

    const int B = in_sizes[0] / HH;          // 2048

    float* td = (float*)d_ws;                // (B, N)
    float* Dd = td + (size_t)B * NN;         // (B, N)

    dim3 g1(B / 16, NN / 16);
    bl_proj_kernel<<<g1, 32, 0, stream>>>(hidden, Wq, bq, Wp, bp, Wo, bo, td, Dd);

    const size_t smem = (size_t)NN * LDA * sizeof(float);   // 66 KB of 320 KB WGP LDS
    bl_solve_kernel<<<B, 256, smem, stream>>>(pi, sigma, td, Dd, mu);
}